// EarthMoversDistance_19980187861535
// MI455X (gfx1250) — compile-verified
//
#include <hip/hip_runtime.h>
#include <math.h>

typedef float v2f __attribute__((ext_vector_type(2)));
typedef float v8f __attribute__((ext_vector_type(8)));

#define SEG 32          // row segments for column (g) pass
#define N_ITERS 300

// ---------------------------------------------------------------- helpers
__device__ __forceinline__ void lse_push(float t, float& m, float& s) {
    if (t > m) { s = s * expf(m - t) + 1.0f; m = t; }
    else       { s += expf(t - m); }
}
__device__ __forceinline__ void lse_merge(float m2, float s2, float& m, float& s) {
    float M = fmaxf(m, m2);
    s = s * expf(m - M) + s2 * expf(m2 - M);
    m = M;
}

// ---------------------------------------------------------------- prep: |x|^2, |y|^2, f=g=0
__global__ void emd_prep_kernel(const float* __restrict__ X, const float* __restrict__ Y,
                                float* __restrict__ x2, float* __restrict__ y2,
                                float* __restrict__ f, float* __restrict__ g, int n) {
    int i = blockIdx.x * blockDim.x + threadIdx.x;
    if (i < n) {
        float a0 = X[i*3+0], a1 = X[i*3+1], a2 = X[i*3+2];
        float b0 = Y[i*3+0], b1 = Y[i*3+1], b2 = Y[i*3+2];
        x2[i] = a0*a0 + a1*a1 + a2*a2;
        y2[i] = b0*b0 + b1*b1 + b2*b2;
        f[i] = 0.0f;
        g[i] = 0.0f;
    }
}

// ---------------------------------------------------------------- cost matrix via WMMA f32 16x16x4
// One wave32 per 16x16 tile of C. Gram term via V_WMMA_F32_16X16X4_F32 (K=3 padded to 4).
__global__ void emd_cost_wmma_kernel(const float* __restrict__ X, const float* __restrict__ Y,
                                     const float* __restrict__ x2, const float* __restrict__ y2,
                                     float* __restrict__ C, float* __restrict__ blockSums, int n) {
    const int lane = threadIdx.x & 31;
    const int wave = threadIdx.x >> 5;
    const int tcols = n >> 4;
    const int tile  = blockIdx.x * 8 + wave;
    const int ti = tile / tcols;
    const int tj = tile % tcols;

    const int h  = lane >> 4;      // lane half: 0 -> K=0,1 ; 1 -> K=2,3
    const int l  = lane & 15;
    const int Mr = ti * 16 + l;    // A row held by this lane
    const int Nc = tj * 16 + l;    // B column held by this lane
    const int k0 = 2 * h;

    v2f a, b;
    a.x = X[Mr*3 + k0];
    a.y = (k0 + 1 < 3) ? X[Mr*3 + k0 + 1] : 0.0f;
    b.x = Y[Nc*3 + k0];
    b.y = (k0 + 1 < 3) ? Y[Nc*3 + k0 + 1] : 0.0f;

    v8f acc = {};
    acc = __builtin_amdgcn_wmma_f32_16x16x4_f32(false, a, false, b, (short)0, acc, false, false);

    const float yj2 = y2[Nc];
    float lsum = 0.0f;
#pragma unroll
    for (int r = 0; r < 8; ++r) {
        int i = ti * 16 + r + 8 * h;              // D layout: vgpr r -> M = r + 8*h
        float d2 = x2[i] + yj2 - 2.0f * acc[r];
        float cv = sqrtf(fmaxf(d2, 0.0f) + 1e-12f);
        C[(size_t)i * n + Nc] = cv;
        lsum += cv;
    }

    __shared__ float red[256];
    red[threadIdx.x] = lsum;
    __syncthreads();
    for (int o = 128; o > 0; o >>= 1) {
        if (threadIdx.x < o) red[threadIdx.x] += red[threadIdx.x + o];
        __syncthreads();
    }
    if (threadIdx.x == 0) blockSums[blockIdx.x] = red[0];
}

// ---------------------------------------------------------------- eps = 0.02 * mean(C)
__global__ void emd_eps_kernel(const float* __restrict__ blockSums, int nb,
                               float* __restrict__ epsOut, float invCount) {
    __shared__ float red[1024];
    float s = 0.0f;
    for (int i = threadIdx.x; i < nb; i += 1024) s += blockSums[i];
    red[threadIdx.x] = s;
    __syncthreads();
    for (int o = 512; o > 0; o >>= 1) {
        if (threadIdx.x < o) red[threadIdx.x] += red[threadIdx.x + o];
        __syncthreads();
    }
    if (threadIdx.x == 0) epsOut[0] = 0.02f * red[0] * invCount;
}

// ---------------------------------------------------------------- f_i = -eps*(log_u + LSE_j((g_j - C_ij)/eps))
// one wave per row; g staged in LDS; coalesced row reads (L2-resident C)
__global__ void emd_f_kernel(const float* __restrict__ C, const float* __restrict__ g,
                             float* __restrict__ f, const float* __restrict__ epsP,
                             int n, float log_u) {
    extern __shared__ float sg[];
    for (int i = threadIdx.x; i < n; i += blockDim.x) sg[i] = g[i];
    __syncthreads();

    const int lane = threadIdx.x & 31;
    const int row  = blockIdx.x * 8 + (threadIdx.x >> 5);
    const float eps = epsP[0];
    const float inv = 1.0f / eps;
    const float* Crow = C + (size_t)row * n;

    float m = -INFINITY, s = 0.0f;
    for (int c = lane; c < n; c += 32) {
        float t = (sg[c] - Crow[c]) * inv;
        lse_push(t, m, s);
    }
#pragma unroll
    for (int o = 16; o > 0; o >>= 1) {
        float m2 = __shfl_xor(m, o, 32);
        float s2 = __shfl_xor(s, o, 32);
        lse_merge(m2, s2, m, s);
    }
    if (lane == 0) f[row] = -eps * (log_u + m + logf(s));
}

// ---------------------------------------------------------------- g pass, stage 1: per-segment partial (m,s)
// wave handles 32 consecutive columns over a 128-row segment; contiguous 128B reads per row
__global__ void emd_g_partial_kernel(const float* __restrict__ C, const float* __restrict__ f,
                                     float* __restrict__ pm, float* __restrict__ ps,
                                     const float* __restrict__ epsP, int n) {
    const int rs = n / SEG;                       // rows per segment
    __shared__ float sf[8 * 128];                 // 8 consecutive segments per block
    const int wid0 = blockIdx.x * 8;
    const int segBase = (wid0 & (SEG - 1)) * rs;  // rows covered by this block's 8 waves
    for (int i = threadIdx.x; i < 8 * rs; i += blockDim.x) sf[i] = f[segBase + i];
    __syncthreads();

    const int lane = threadIdx.x & 31;
    const int wid  = wid0 + (threadIdx.x >> 5);
    const int seg  = wid & (SEG - 1);
    const int col  = (wid >> 5) * 32 + lane;
    const float inv = 1.0f / epsP[0];

    const int r0 = seg * rs;
    float m = -INFINITY, s = 0.0f;
    for (int r = 0; r < rs; ++r) {
        float t = (sf[(r0 - segBase) + r] - C[(size_t)(r0 + r) * n + col]) * inv;
        lse_push(t, m, s);
    }
    pm[(size_t)seg * n + col] = m;
    ps[(size_t)seg * n + col] = s;
}

// ---------------------------------------------------------------- g pass, stage 2: combine segments
__global__ void emd_g_combine_kernel(const float* __restrict__ pm, const float* __restrict__ ps,
                                     float* __restrict__ g, const float* __restrict__ epsP,
                                     int n, float log_u) {
    const int col = blockIdx.x * blockDim.x + threadIdx.x;
    if (col >= n) return;
    const float eps = epsP[0];
    float m = -INFINITY, s = 0.0f;
    for (int seg = 0; seg < SEG; ++seg)
        lse_merge(pm[(size_t)seg * n + col], ps[(size_t)seg * n + col], m, s);
    g[col] = -eps * (log_u + m + logf(s));
}

// ---------------------------------------------------------------- EMD = sum_ij exp(2log_u + (f+g-C)/eps) * C
__global__ void emd_sum_partial_kernel(const float* __restrict__ C, const float* __restrict__ f,
                                       const float* __restrict__ g, const float* __restrict__ epsP,
                                       float* __restrict__ blockSums, int n, float log_u) {
    extern __shared__ float sgm[];               // [n] staged g + [256] reduction
    float* sg   = sgm;
    float* sred = sgm + n;
    for (int i = threadIdx.x; i < n; i += blockDim.x) sg[i] = g[i];
    __syncthreads();

    const int lane = threadIdx.x & 31;
    const int row  = blockIdx.x * 8 + (threadIdx.x >> 5);
    const float inv = 1.0f / epsP[0];
    const float fi  = f[row];
    const float* Crow = C + (size_t)row * n;

    float s = 0.0f;
    for (int c = lane; c < n; c += 32) {
        float cv = Crow[c];
        s += expf(2.0f * log_u + (fi + sg[c] - cv) * inv) * cv;
    }
    sred[threadIdx.x] = s;
    __syncthreads();
    for (int o = 128; o > 0; o >>= 1) {
        if (threadIdx.x < o) sred[threadIdx.x] += sred[threadIdx.x + o];
        __syncthreads();
    }
    if (threadIdx.x == 0) blockSums[blockIdx.x] = sred[0];
}

__global__ void emd_sum_final_kernel(const float* __restrict__ blockSums, int nb,
                                     float* __restrict__ out) {
    __shared__ float red[512];
    float s = 0.0f;
    for (int i = threadIdx.x; i < nb; i += 512) s += blockSums[i];
    red[threadIdx.x] = s;
    __syncthreads();
    for (int o = 256; o > 0; o >>= 1) {
        if (threadIdx.x < o) red[threadIdx.x] += red[threadIdx.x + o];
        __syncthreads();
    }
    if (threadIdx.x == 0) out[0] = red[0];
}

// ---------------------------------------------------------------- launch
extern "C" void kernel_launch(void* const* d_in, const int* in_sizes, int n_in,
                              void* d_out, int out_size, void* d_ws, size_t ws_size,
                              hipStream_t stream) {
    const float* X = (const float*)d_in[0];     // pc1 [n,3] fp32
    const float* Y = (const float*)d_in[1];     // pc2 [n,3] fp32
    float* out = (float*)d_out;
    const int n = in_sizes[0] / 3;              // 4096
    const float log_u = -logf((float)n);

    // workspace layout (floats)
    float* W = (float*)d_ws;
    size_t off = 0;
    float* C  = W + off; off += (size_t)n * n;   // 64 MB: resident in 192MB L2
    float* x2 = W + off; off += n;
    float* y2 = W + off; off += n;
    float* f  = W + off; off += n;
    float* g  = W + off; off += n;
    float* pm = W + off; off += (size_t)SEG * n;
    float* ps = W + off; off += (size_t)SEG * n;
    const int costBlocks = (n / 16) * (n / 16) / 8;   // 8 waves (tiles) per 256-thread block
    float* bsum = W + off; off += costBlocks;
    const int rowBlocks = n / 8;                      // 8 rows (waves) per block
    float* esum = W + off; off += rowBlocks;
    float* epsP = W + off; off += 1;

    // 1) prep
    emd_prep_kernel<<<(n + 255) / 256, 256, 0, stream>>>(X, Y, x2, y2, f, g, n);
    // 2) cost matrix via WMMA + total for eps
    emd_cost_wmma_kernel<<<costBlocks, 256, 0, stream>>>(X, Y, x2, y2, C, bsum, n);
    emd_eps_kernel<<<1, 1024, 0, stream>>>(bsum, costBlocks, epsP, 1.0f / ((float)n * (float)n));

    // 3) Sinkhorn iterations
    const int gPartBlocks = (n / 32) * SEG / 8;       // 8 waves per block
    const size_t shRow = (size_t)n * sizeof(float);
    for (int it = 0; it < N_ITERS; ++it) {
        emd_f_kernel<<<rowBlocks, 256, shRow, stream>>>(C, g, f, epsP, n, log_u);
        emd_g_partial_kernel<<<gPartBlocks, 256, 0, stream>>>(C, f, pm, ps, epsP, n);
        emd_g_combine_kernel<<<(n + 255) / 256, 256, 0, stream>>>(pm, ps, g, epsP, n, log_u);
    }

    // 4) EMD = sum(P * C)
    emd_sum_partial_kernel<<<rowBlocks, 256, shRow + 256 * sizeof(float), stream>>>(
        C, f, g, epsP, esum, n, log_u);
    emd_sum_final_kernel<<<1, 512, 0, stream>>>(esum, rowBlocks, out);
}